// InjectedEncoder_40089224741457
// MI455X (gfx1250) — compile-verified
//
#include <hip/hip_runtime.h>
#include <math.h>

typedef float v2f __attribute__((ext_vector_type(2)));
typedef float v8f __attribute__((ext_vector_type(8)));

#define WMMA_F32X4(a,b,c) \
  __builtin_amdgcn_wmma_f32_16x16x4_f32(false,(a),false,(b),(short)0,(c),false,false)

__device__ __forceinline__ v8f zero8() {
  v8f v = {0.f,0.f,0.f,0.f,0.f,0.f,0.f,0.f};
  return v;
}

// ---------------- problem constants ----------------
constexpr int B_   = 16;
constexpr int S_   = 64;
constexpr int NF   = 64;
constexpr int NC   = 8;
constexpr int LATN = 512;
constexpr int CIN  = 3;
constexpr int CW   = 24;
constexpr int PIX  = S_ * S_;          // 4096
constexpr int FLAT = 12481;
constexpr int SPAD = 12484;            // FLAT padded to multiple of 4 (zero tail)

// per-batch hyper-parameter layout (weight row-major [Cout][Cin], then bias)
constexpr int O_BT  = 0;        // 64x1024 + 64  = 65600
constexpr int O_C2W = 65600;    // 24x64  + 24   = 1560
constexpr int O_W2C = 67160;    // 128x152+ 128  = 19584
constexpr int O_G2C = 86744;    // 128x128+ 128  = 16512
constexpr int O_CC1 = 103256;   // 128x256+ 128  = 32896
constexpr int O_CC2 = 136152;   // 64x128 + 64   = 8256
constexpr int O_CCS = 144408;   // 64x256 + 64   = 16448
constexpr int PTOT  = 160856;

// ---------------- MT19937 + legacy numpy gauss: _GKS[0..7] (8 x 2 x 3x3) ----------------
__global__ void init_gk_kernel(float* __restrict__ gk) {
  if (threadIdx.x != 0 || blockIdx.x != 0) return;
  unsigned mt[624];
  // init_genrand(19650218)
  mt[0] = 19650218u;
  for (int i = 1; i < 624; ++i)
    mt[i] = 1812433253u * (mt[i-1] ^ (mt[i-1] >> 30)) + (unsigned)i;
  // init_by_array(key=[0])
  {
    int i = 1, j = 0;
    for (int k = 624; k; --k) {
      mt[i] = (mt[i] ^ ((mt[i-1] ^ (mt[i-1] >> 30)) * 1664525u)) + 0u + (unsigned)j;
      ++i; ++j;
      if (i >= 624) { mt[0] = mt[623]; i = 1; }
      if (j >= 1) j = 0;
    }
    for (int k = 623; k; --k) {
      mt[i] = (mt[i] ^ ((mt[i-1] ^ (mt[i-1] >> 30)) * 1566083941u)) - (unsigned)i;
      ++i;
      if (i >= 624) { mt[0] = mt[623]; i = 1; }
    }
    mt[0] = 0x80000000u;
  }
  int mti = 624;
  auto next = [&]() -> unsigned {
    if (mti >= 624) {
      for (int kk = 0; kk < 624; ++kk) {
        unsigned y = (mt[kk] & 0x80000000u) | (mt[(kk+1) % 624] & 0x7fffffffu);
        mt[kk] = mt[(kk+397) % 624] ^ (y >> 1) ^ ((y & 1u) ? 2567483615u : 0u);
      }
      mti = 0;
    }
    unsigned y = mt[mti++];
    y ^= y >> 11; y ^= (y << 7) & 2636928640u; y ^= (y << 15) & 4022730752u; y ^= y >> 18;
    return y;
  };
  auto rdouble = [&]() -> double {
    unsigned a = next() >> 5, b = next() >> 6;
    return ((double)a * 67108864.0 + (double)b) / 9007199254740992.0;
  };
  double cache = 0.0; bool has = false;
  auto gauss = [&]() -> double {
    if (has) { has = false; return cache; }
    double x1, x2, r2;
    do {
      x1 = 2.0 * rdouble() - 1.0;
      x2 = 2.0 * rdouble() - 1.0;
      r2 = x1*x1 + x2*x2;
    } while (r2 >= 1.0 || r2 == 0.0);
    double f = sqrt(-2.0 * log(r2) / r2);
    cache = f * x1; has = true;
    return f * x2;
  };
  for (int n = 0; n < 144; ++n) {
    float g = (float)gauss();           // randn -> astype(f32)
    gk[n] = g / 9.0f;                   // / (s*s), s=3
  }
}

// ---------------- WMMA GEMM: hyper-weight generation p = 0.01 * lat @ hw ----------------
// lat: (16,512) row-major; hw: (512,N) row-major; pOut[b*PTOT + off + j]
__global__ __launch_bounds__(32)
void hyper_gemm_wmma(const float* __restrict__ lat, const float* __restrict__ hw,
                     int N, float* __restrict__ pOut, int off) {
  const int lane = threadIdx.x;
  const int ln = lane & 15, hi = lane >> 4;
  const int n0 = blockIdx.x * 16;
  int colc = n0 + ln; if (colc > N - 1) colc = N - 1;   // clamp loads on ragged edge
  v8f acc = zero8();
  for (int k0 = 0; k0 < LATN; k0 += 4) {
    v2f av = *(const v2f*)(lat + (long)ln * LATN + k0 + 2*hi);
    const float* hp = hw + (long)(k0 + 2*hi) * N + colc;
    v2f bv; bv.x = hp[0]; bv.y = hp[N];
    acc = WMMA_F32X4(av, bv, acc);
  }
  const int c = n0 + ln;
  if (c < N) {
#pragma unroll
    for (int r = 0; r < 8; ++r) {
      int b = r + 8*hi;                         // M index = batch
      pOut[(long)b * PTOT + off + c] = 0.01f * acc[r];
    }
  }
}

// ---------------- WMMA GEMM over pixels: Y[b, yOff+m, p] (= / relu / out+=leak*(..+Z)) ----
// P: hyper params base; weights [Cout][Cin] at wOff, bias at wOff+Cout*Cin (per batch)
// X: (B, Cin, 4096) with batch stride xStride (0 = broadcast)
// mode 0: Y = W@X + b ; mode 1: relu ; mode 2: Y += leak*(W@X + b + Z), Z:(B,Cout,4096)
__global__ __launch_bounds__(32)
void gemm_px_wmma(const float* __restrict__ P, int wOff, int Cout, int Cin,
                  const float* __restrict__ X, long xStride,
                  float* __restrict__ Y, int yCtot, int yOff,
                  int mode, const float* __restrict__ Z,
                  const float* __restrict__ leakPtr) {
  const int lane = threadIdx.x;
  const int ln = lane & 15, hi = lane >> 4;
  const int b  = blockIdx.z;
  const int m0 = blockIdx.y * 16;
  const int n0 = blockIdx.x * 64;                 // 4 N-tiles per wave
  const float* W    = P + (long)b * PTOT + wOff;
  const float* bias = W + (long)Cout * Cin;
  const float* Xb   = X + (long)b * xStride;
  const int arow = m0 + ln;

  v8f acc[4];
#pragma unroll
  for (int t = 0; t < 4; ++t) acc[t] = zero8();

  for (int k0 = 0; k0 < Cin; k0 += 4) {
    v2f av = *(const v2f*)(W + (long)arow * Cin + k0 + 2*hi);
    const float* xp = Xb + (long)(k0 + 2*hi) * PIX + n0 + ln;
#pragma unroll
    for (int t = 0; t < 4; ++t) {
      v2f bv; bv.x = xp[t*16]; bv.y = xp[PIX + t*16];
      acc[t] = WMMA_F32X4(av, bv, acc[t]);
    }
  }

  float leak = 0.f;
  if (mode == 2) {
    float lf = leakPtr[0];
    leak = fminf(fmaxf(lf, 0.001f), 1000.0f);
  }
#pragma unroll
  for (int r = 0; r < 8; ++r) {
    int m = m0 + r + 8*hi;
    float bv = bias[m];
#pragma unroll
    for (int t = 0; t < 4; ++t) {
      long col = n0 + t*16 + ln;
      long yi = ((long)b * yCtot + yOff + m) * PIX + col;
      float v = acc[t][r] + bv;
      if (mode == 0)      Y[yi] = v;
      else if (mode == 1) Y[yi] = fmaxf(v, 0.f);
      else {
        long zi = ((long)b * Cout + m) * PIX + col;
        Y[yi] += leak * (v + Z[zi]);
      }
    }
  }
}

// ---------------- WMMA GEMM: lat_out = state @ out_lat_w + out_lat_b ----------------
__global__ __launch_bounds__(32)
void outlat_wmma(const float* __restrict__ st, const float* __restrict__ W,
                 const float* __restrict__ bias, float* __restrict__ latOut) {
  const int lane = threadIdx.x;
  const int ln = lane & 15, hi = lane >> 4;
  const int n0 = blockIdx.x * 16;
  v8f acc = zero8();
  for (int k0 = 0; k0 < SPAD; k0 += 4) {
    v2f av = *(const v2f*)(st + (long)ln * SPAD + k0 + 2*hi);
    int r0 = k0 + 2*hi, r1 = r0 + 1;
    if (r0 > FLAT - 1) r0 = FLAT - 1;     // tail rows multiply state==0 anyway
    if (r1 > FLAT - 1) r1 = FLAT - 1;
    v2f bv; bv.x = W[(long)r0 * LATN + n0 + ln]; bv.y = W[(long)r1 * LATN + n0 + ln];
    acc = WMMA_F32X4(av, bv, acc);
  }
#pragma unroll
  for (int r = 0; r < 8; ++r)
    latOut[(long)(r + 8*hi) * LATN + n0 + ln] = acc[r] + bias[n0 + ln];
}

// ---------------- elementwise / reduction kernels ----------------
__global__ __launch_bounds__(256)
void conv_img_kernel(const float* __restrict__ x, const float* __restrict__ w,
                     const float* __restrict__ bias, float* __restrict__ out) {
  int b = blockIdx.z, o = blockIdx.y;
  int pix = blockIdx.x * 256 + threadIdx.x;
  float s = bias[o];
#pragma unroll
  for (int i = 0; i < CIN; ++i)
    s += w[o*CIN + i] * x[((long)b*CIN + i) * PIX + pix];
  out[((long)b*NF + o) * PIX + pix] = s;
}

__global__ __launch_bounds__(256)
void waveb_kernel(const float* __restrict__ surf, float* __restrict__ xcat) {
  int b = blockIdx.z, j = blockIdx.y;
  int pix = blockIdx.x * 256 + threadIdx.x;
  float sv = surf[((long)b*NF + j) * PIX + pix];
  const float E = 2.71828182845904523536f;
  xcat[((long)b*152 + 24 + j) * PIX + pix] = E * cosf(sv);
  xcat[((long)b*152 + 88 + j) * PIX + pix] = E * sinf(sv);
}

__global__ __launch_bounds__(64)
void meanwh_kernel(const float* __restrict__ out, float* __restrict__ mw,
                   float* __restrict__ mh, float* __restrict__ omp) {
  int b = blockIdx.y, o = blockIdx.x, t = threadIdx.x;
  const float* p = out + ((long)b*NF + o) * PIX;
  float rs = 0.f, cs = 0.f;
  for (int k = 0; k < 64; ++k) { rs += p[t*64 + k]; cs += p[k*64 + t]; }
  mw[((long)b*NF + o)*64 + t] = rs * (1.f/64.f);
  mh[((long)b*NF + o)*64 + t] = cs * (1.f/64.f);
  __shared__ float sh[64];
  sh[t] = rs; __syncthreads();
  for (int off = 32; off > 0; off >>= 1) { if (t < off) sh[t] += sh[t + off]; __syncthreads(); }
  if (t == 0) omp[b*NF + o] = sh[0] * (1.f/4096.f);
}

__global__ void wavemean_kernel(const float* __restrict__ p, const float* __restrict__ omp,
                                float* __restrict__ wm) {
  int idx = blockIdx.x * blockDim.x + threadIdx.x;
  if (idx >= B_ * CW) return;
  int b = idx / CW, j = idx % CW;
  const float* W = p + (long)b * PTOT + O_C2W;
  float s = W[CW*NF + j];                       // bias
  for (int i = 0; i < NF; ++i) s += W[j*NF + i] * omp[b*NF + i];
  wm[idx] = s;
}

__global__ __launch_bounds__(256)
void xcatwave_kernel(const float* __restrict__ wm, int c, float* __restrict__ xcat) {
  int b = blockIdx.z, j = blockIdx.y;          // j < 24
  int pix = blockIdx.x * 256 + threadIdx.x;
  int k = j >> 2, t = j & 3;
  double pos = (double)((t < 2) ? (pix & 63) : (pix >> 6)) / 64.0;
  double ang = (double)(1 << k) * 3.14159265358979323846 * pos
             + 2.0 * 3.14159265358979323846 * (double)c / 8.0;
  double val = (t & 1) ? cos(ang) : sin(ang);
  xcat[((long)b*152 + j) * PIX + pix] = wm[b*CW + j] * (float)val;
}

__global__ __launch_bounds__(256)
void gradconv_kernel(const float* __restrict__ out, const float* __restrict__ gk,
                     int c, float* __restrict__ g) {
  int b = blockIdx.z, i = blockIdx.y;
  int pix = blockIdx.x * 256 + threadIdx.x;
  int y = pix >> 6, x = pix & 63;
  const float* kk = gk + c * 18;
  const float* src = out + ((long)b*NF + i) * PIX;
  float s0 = 0.f, s1 = 0.f;
#pragma unroll
  for (int dy = 0; dy < 3; ++dy) {
    int yy = y + dy - 1;
    if ((unsigned)yy >= 64u) continue;
#pragma unroll
    for (int dx = 0; dx < 3; ++dx) {
      int xx = x + dx - 1;
      if ((unsigned)xx >= 64u) continue;
      float v = src[yy*64 + xx];
      s0 += v * kk[dy*3 + dx];
      s1 += v * kk[9 + dy*3 + dx];
    }
  }
  g[((long)b*128 + 2*i)     * PIX + pix] = s0;
  g[((long)b*128 + 2*i + 1) * PIX + pix] = s1;
}

__global__ __launch_bounds__(256)
void instnorm_kernel(float* __restrict__ X) {   // (B, 256, 4096) in place
  int b = blockIdx.y, ch = blockIdx.x;
  float* p = X + ((long)b*256 + ch) * PIX;
  __shared__ float ss[256], ss2[256];
  float s = 0.f, s2 = 0.f;
  for (int i = threadIdx.x; i < PIX; i += 256) { float v = p[i]; s += v; s2 += v*v; }
  ss[threadIdx.x] = s; ss2[threadIdx.x] = s2; __syncthreads();
  for (int off = 128; off > 0; off >>= 1) {
    if (threadIdx.x < off) { ss[threadIdx.x] += ss[threadIdx.x+off]; ss2[threadIdx.x] += ss2[threadIdx.x+off]; }
    __syncthreads();
  }
  float m  = ss[0] * (1.f/4096.f);
  float var = ss2[0] * (1.f/4096.f) - m*m;
  float inv = rsqrtf(var + 1e-5f);
  for (int i = threadIdx.x; i < PIX; i += 256) p[i] = (p[i] - m) * inv;
}

__global__ void prepv_kernel(const float* __restrict__ Wc, const float* __restrict__ bc,
                             float* __restrict__ vbuf) {
  int i = threadIdx.x;                          // 64 threads
  float s1 = 0.f, s2 = 0.f, s5 = 0.f, s6 = 0.f;
  for (int o = 0; o < 64; ++o) {
    s1 += Wc[(64  + o)*NF + i];
    s2 += Wc[(128 + o)*NF + i];
    s5 += Wc[(320 + o)*NF + i];
    s6 += Wc[(384 + o)*NF + i];
  }
  vbuf[i]       = s1 * (1.f/64.f);
  vbuf[64 + i]  = s2 * (1.f/64.f);
  vbuf[128 + i] = s5 * (1.f/64.f);
  vbuf[192 + i] = s6 * (1.f/64.f);
  if (i == 0) {
    float c1 = 0.f, c2 = 0.f, c5 = 0.f, c6 = 0.f;
    for (int o = 0; o < 64; ++o) { c1 += bc[64+o]; c2 += bc[128+o]; c5 += bc[320+o]; c6 += bc[384+o]; }
    vbuf[256] = c1*(1.f/64.f); vbuf[257] = c2*(1.f/64.f);
    vbuf[258] = c5*(1.f/64.f); vbuf[259] = c6*(1.f/64.f);
  }
}

__global__ __launch_bounds__(256)
void state_kernel(const float* __restrict__ out, const float* __restrict__ omp,
                  const float* __restrict__ mw, const float* __restrict__ mh,
                  const float* __restrict__ vbuf, const float* __restrict__ Wc,
                  const float* __restrict__ bc, float* __restrict__ st) {
  int b = blockIdx.y;
  int f = blockIdx.x * 256 + threadIdx.x;
  if (f >= SPAD) return;
  float r = 0.f;
  if (f < 64) {                                  // parts[0].mean((2,3))
    r = bc[f];
    for (int i = 0; i < NF; ++i) r += Wc[f*NF + i] * omp[b*NF + i];
  } else if (f < 128) {                          // parts[1].mean((1,3)) -> per h
    int h = f - 64; r = vbuf[256];
    for (int i = 0; i < NF; ++i) r += vbuf[i] * mw[((long)b*NF + i)*64 + h];
  } else if (f < 192) {                          // parts[2].mean((1,2)) -> per w
    int w = f - 128; r = vbuf[257];
    for (int i = 0; i < NF; ++i) r += vbuf[64 + i] * mh[((long)b*NF + i)*64 + w];
  } else if (f < 4288) {                         // parts[3].mean(3) -> (o,h)
    int idx = f - 192; int o = 192 + (idx >> 6), h = idx & 63;
    r = bc[o];
    for (int i = 0; i < NF; ++i) r += Wc[o*NF + i] * mw[((long)b*NF + i)*64 + h];
  } else if (f < 8384) {                         // parts[4].mean(2) -> (o,w)
    int idx = f - 4288; int o = 256 + (idx >> 6), w = idx & 63;
    r = bc[o];
    for (int i = 0; i < NF; ++i) r += Wc[o*NF + i] * mh[((long)b*NF + i)*64 + w];
  } else if (f < 12480) {                        // parts[5].mean(1) -> per pixel
    int pix = f - 8384; r = vbuf[258];
    for (int i = 0; i < NF; ++i) r += vbuf[128 + i] * out[((long)b*NF + i)*PIX + pix];
  } else if (f == 12480) {                       // parts[6].mean((1,2,3))
    r = vbuf[259];
    for (int i = 0; i < NF; ++i) r += vbuf[192 + i] * omp[b*NF + i];
  }                                              // f in [12481,12484): zero pad
  st[(long)b * SPAD + f] = r;
}

// ---------------- host launcher ----------------
extern "C" void kernel_launch(void* const* d_in, const int* in_sizes, int n_in,
                              void* d_out, int out_size, void* d_ws, size_t ws_size,
                              hipStream_t stream) {
  (void)in_sizes; (void)n_in; (void)out_size; (void)ws_size;

  const float* x        = (const float*)d_in[0];
  const float* inj_lat  = (const float*)d_in[1];
  const float* cimg_w   = (const float*)d_in[2];
  const float* cimg_b   = (const float*)d_in[3];
  const float* wave_bias= (const float*)d_in[4];
  const float* hw_bt    = (const float*)d_in[5];
  const float* hw_c2w   = (const float*)d_in[6];
  const float* hw_w2c   = (const float*)d_in[7];
  const float* hw_g2c   = (const float*)d_in[8];
  const float* hw_cc1   = (const float*)d_in[9];
  const float* hw_cc2   = (const float*)d_in[10];
  const float* hw_ccs   = (const float*)d_in[11];
  const float* ocw      = (const float*)d_in[12];
  const float* ocb      = (const float*)d_in[13];
  const float* olw      = (const float*)d_in[14];
  const float* olb      = (const float*)d_in[15];
  const float* leak     = (const float*)d_in[16];

  float* latOut = (float*)d_out;                 // (B, 512)
  float* outBuf = (float*)d_out + B_ * LATN;     // (B, 64, 4096) 'out' built in place

  // workspace carve-out (~186 MB of floats), 256B-aligned chunks
  float* ws = (float*)d_ws;
  size_t cur = 0;
  auto alloc = [&](size_t n) { float* p = ws + cur; cur += (n + 63) & ~(size_t)63; return p; };
  float* gk     = alloc(160);
  float* vbuf   = alloc(272);
  float* wm     = alloc((size_t)B_ * CW);
  float* omp    = alloc((size_t)B_ * NF);
  float* mw     = alloc((size_t)B_ * NF * 64);
  float* mh     = alloc((size_t)B_ * NF * 64);
  float* stateB = alloc((size_t)B_ * SPAD);
  float* pAll   = alloc((size_t)B_ * PTOT);
  float* xcat   = alloc((size_t)B_ * 152 * PIX);  // ch 0..23 wave_c, 24..151 wave_b (loop-invariant)
  float* gbuf   = alloc((size_t)B_ * 128 * PIX);  // depthwise grads; reused as cc2 output t1
  float* onew   = alloc((size_t)B_ * 256 * PIX);  // [grads_enc | wave_enc]
  float* hbuf   = alloc((size_t)B_ * 128 * PIX);  // surface (pre-loop), cc1 activations (in loop)

  init_gk_kernel<<<dim3(1), dim3(1), 0, stream>>>(gk);
  prepv_kernel<<<dim3(1), dim3(64), 0, stream>>>(ocw, ocb, vbuf);

  // hyper-weight generation (cell-invariant, hoisted): p = 0.01 * inj_lat @ hw
  hyper_gemm_wmma<<<dim3(65600/16), dim3(32), 0, stream>>>(inj_lat, hw_bt,  65600, pAll, O_BT);
  hyper_gemm_wmma<<<dim3((1560+15)/16), dim3(32), 0, stream>>>(inj_lat, hw_c2w, 1560, pAll, O_C2W);
  hyper_gemm_wmma<<<dim3(19584/16), dim3(32), 0, stream>>>(inj_lat, hw_w2c, 19584, pAll, O_W2C);
  hyper_gemm_wmma<<<dim3(16512/16), dim3(32), 0, stream>>>(inj_lat, hw_g2c, 16512, pAll, O_G2C);
  hyper_gemm_wmma<<<dim3(32896/16), dim3(32), 0, stream>>>(inj_lat, hw_cc1, 32896, pAll, O_CC1);
  hyper_gemm_wmma<<<dim3(8256/16),  dim3(32), 0, stream>>>(inj_lat, hw_cc2, 8256,  pAll, O_CC2);
  hyper_gemma_dummy: ;
  hyper_gemm_wmma<<<dim3(16448/16), dim3(32), 0, stream>>>(inj_lat, hw_ccs, 16448, pAll, O_CCS);

  // out = conv_img(x)
  conv_img_kernel<<<dim3(16, NF, B_), dim3(256), 0, stream>>>(x, cimg_w, cimg_b, outBuf);

  // surface = dconv(broadcast(wave_bias), bias_trans) -> hbuf ; wave_b -> xcat[:,24:152]
  gemm_px_wmma<<<dim3(64, 4, B_), dim3(32), 0, stream>>>(
      pAll, O_BT, 64, 1024, wave_bias, 0L, hbuf, 64, 0, 0, nullptr, nullptr);
  waveb_kernel<<<dim3(16, NF, B_), dim3(256), 0, stream>>>(hbuf, xcat);

  for (int c = 0; c < NC; ++c) {
    meanwh_kernel<<<dim3(NF, B_), dim3(64), 0, stream>>>(outBuf, mw, mh, omp);
    wavemean_kernel<<<dim3(2), dim3(256), 0, stream>>>(pAll, omp, wm);
    xcatwave_kernel<<<dim3(16, CW, B_), dim3(256), 0, stream>>>(wm, c, xcat);
    // wave_enc -> onew[:,128:256]
    gemm_px_wmma<<<dim3(64, 8, B_), dim3(32), 0, stream>>>(
        pAll, O_W2C, 128, 152, xcat, (long)152*PIX, onew, 256, 128, 0, nullptr, nullptr);
    // depthwise 3x3 grads
    gradconv_kernel<<<dim3(16, NF, B_), dim3(256), 0, stream>>>(outBuf, gk, c, gbuf);
    // grads_enc -> onew[:,0:128]
    gemm_px_wmma<<<dim3(64, 8, B_), dim3(32), 0, stream>>>(
        pAll, O_G2C, 128, 128, gbuf, (long)128*PIX, onew, 256, 0, 0, nullptr, nullptr);
    instnorm_kernel<<<dim3(256, B_), dim3(256), 0, stream>>>(onew);
    // h = relu(cc1 @ onew)
    gemm_px_wmma<<<dim3(64, 8, B_), dim3(32), 0, stream>>>(
        pAll, O_CC1, 128, 256, onew, (long)256*PIX, hbuf, 128, 0, 1, nullptr, nullptr);
    // t1 = cc2 @ h   (reuses gbuf)
    gemm_px_wmma<<<dim3(64, 4, B_), dim3(32), 0, stream>>>(
        pAll, O_CC2, 64, 128, hbuf, (long)128*PIX, gbuf, 64, 0, 0, nullptr, nullptr);
    // out += leak * (ccs @ onew + bias + t1)
    gemm_px_wmma<<<dim3(64, 4, B_), dim3(32), 0, stream>>>(
        pAll, O_CCS, 64, 256, onew, (long)256*PIX, outBuf, 64, 0, 2, gbuf, leak);
  }

  // final readout: means pushed through out_conv_w (cs never materialized)
  meanwh_kernel<<<dim3(NF, B_), dim3(64), 0, stream>>>(outBuf, mw, mh, omp);
  state_kernel<<<dim3((SPAD + 255)/256, B_), dim3(256), 0, stream>>>(
      outBuf, omp, mw, mh, vbuf, ocw, ocb, stateB);
  outlat_wmma<<<dim3(LATN/16), dim3(32), 0, stream>>>(stateB, olw, olb, latOut);
}